// SpaPromptGenBlock_36009005809797
// MI455X (gfx1250) — compile-verified
//
#include <hip/hip_runtime.h>
#include <math.h>
#include <stdint.h>

// ---------------------------------------------------------------------------
// MI455X (gfx1250, wave32).
// Heavy math: fp32 WMMA 16x16x4  D(16x16) = A(16x4) x B(4x16) + C.
// Expert GEMMs run on top-2 compacted token lists (identical math to the
// dense reference since dropped tokens have gate == 0) -> 2x fewer FLOPs.
// Weight staging: Tensor Data Mover (tensor_load_to_lds) with OOB-zero padding
// of HD=170 -> 176, double-buffered and overlapped with WMMA via TENSORcnt.
// Fragment layouts per CDNA5 ISA 7.12.2:
//   A (16x4):  lane l: m = l&15, ks = (l<16)?0:2 ; v[0]=A[m][ks], v[1]=A[m][ks+1]
//   B (4x16):  lane l: n = l&15, ks = (l<16)?0:2 ; v[0]=B[ks][n], v[1]=B[ks+1][n]
//   C/D (16x16): lane l: n = l&15, mb=(l<16)?0:8 ; v[i] = D[mb+i][n]
// ---------------------------------------------------------------------------

typedef float v2f __attribute__((ext_vector_type(2)));
typedef float v8f __attribute__((ext_vector_type(8)));
typedef unsigned int v4u __attribute__((ext_vector_type(4)));
typedef int v4i __attribute__((ext_vector_type(4)));
typedef int v8i __attribute__((ext_vector_type(8)));

#if defined(__has_builtin)
#if __has_builtin(__builtin_amdgcn_tensor_load_to_lds) && \
    __has_builtin(__builtin_amdgcn_s_wait_tensorcnt)
#define HAVE_TDM 1
#endif
#endif
#ifndef HAVE_TDM
#define HAVE_TDM 0
#endif

#define B_   64
#define C_   64
#define HW_  4096
#define P_   5
#define E_   4
#define HD_  170
#define HDP_ 176
#define ST_  128     // tokens per workgroup tile
#define XS_  68      // LDS stride for [ST][C] token-major tiles (bank pad)
#define HS_  180     // LDS stride for H1 [ST][HDP]

// LDS layout (float offsets); ~287 KB dynamic LDS (WGP has 320 KB)
#define OFF_XR   0                         // X_r [128][68] raw x, token-major
#define OFF_OR   (OFF_XR + ST_*XS_)        // Out_r [128][68] conv output
#define OFF_WU   (OFF_OR + ST_*XS_)        // buffer U: W1[c][176]
#define OFF_WV   (OFF_WU + C_*HDP_)        // buffer V: WcT(64x64) / W2[176][64]
#define OFF_H1   (OFF_WV + HDP_*C_)        // H1 [128][180] (compacted rows)
#define OFF_Y    (OFF_H1 + ST_*HS_)        // Y accumulator [128][68]
#define OFF_PR   (OFF_Y  + ST_*XS_)        // prompt [128]
#define OFF_G    (OFF_PR + ST_)            // gates [4][128]
#define OFF_WG   (OFF_G  + E_*ST_)         // w_gate [64][4]
#define OFF_B1   (OFF_WG + 256)            // b1 padded [176]
#define OFF_B2   (OFF_B1 + HDP_)           // b2 [64]
#define OFF_CNT  (OFF_B2 + 64)             // int cnt[4]
#define OFF_LST  (OFF_CNT + 4)             // int list[4][128]
#define LDS_FLOATS (OFF_LST + E_*ST_)
#define LDS_BYTES  (LDS_FLOATS * 4)

// ws layout (floats)
#define WS_EMB   0                         // [64][64]
#define WS_PW    4096                      // [64][5]
#define WS_PART  4416                      // per-block importance[4],load[4] x 2048
#define NBLK     (B_ * (HW_ / ST_))        // 2048

__device__ __forceinline__ v8f wmma4(v2f a, v2f b, v8f c) {
  return __builtin_amdgcn_wmma_f32_16x16x4_f32(false, a, false, b, (short)0, c,
                                               false, false);
}

__device__ __forceinline__ float gelu_tanh(float v) {
  float u  = 0.7978845608028654f * (v + 0.044715f * v * v * v);
  float t  = __expf(-2.0f * fabsf(u));
  float th = (1.0f - t) / (1.0f + t);
  th = (u >= 0.0f) ? th : -th;
  return 0.5f * v * (1.0f + th);
}

#if HAVE_TDM
// 2D tile load global->LDS via the Tensor Data Mover.
// Element size fp32. Reads past (tensor_d0, tensor_d1) return zero, which
// implements the HD=170 -> 176 zero padding in hardware.
// This toolchain's builtin takes 6 args:
//   (uint32x4 g0, int32x8 g1, int32x4 g2, int32x4 g3, int32x8 extra, i32 cpol)
__device__ __forceinline__ void tdm_load_2d(unsigned lds_off_bytes,
                                            const float* gptr,
                                            unsigned tensor_d0,
                                            unsigned tensor_d1,
                                            unsigned tile_d0, unsigned tile_d1,
                                            unsigned stride0_elems) {
  unsigned long long ga = (unsigned long long)(uintptr_t)gptr;
  v4u g0;
  g0[0] = 1u;                                    // count=1, user mode
  g0[1] = lds_off_bytes;                         // lds_addr
  g0[2] = (unsigned)ga;                          // global_addr[31:0]
  g0[3] = (unsigned)((ga >> 32) & 0x1FFFFFFu) | (2u << 30);  // addr[56:32]|type=2
  v8i g1;
  g1[0] = (int)(2u << 16);                       // data_size=4B; no pad/iter/mask
  g1[1] = (int)(tensor_d0 << 16);                // tensor_dim0[15:0]
  g1[2] = (int)((tensor_d0 >> 16) | (tensor_d1 << 16));
  g1[3] = (int)((tensor_d1 >> 16) | (tile_d0 << 16));
  g1[4] = (int)(tile_d1 & 0xFFFFu);              // tile_dim1; tile_dim2=0 (2D)
  g1[5] = (int)stride0_elems;                    // tensor_dim0_stride[31:0]
  g1[6] = 0;
  g1[7] = 0;
  v4i g2; g2[0] = 1; g2[1] = 1; g2[2] = 0; g2[3] = 0;   // dim2/dim3 benign
  v4i g3; g3[0] = 0; g3[1] = 0; g3[2] = 0; g3[3] = 0;
  v8i gx; gx[0] = 0; gx[1] = 0; gx[2] = 0; gx[3] = 0;
          gx[4] = 0; gx[5] = 0; gx[6] = 0; gx[7] = 0;   // unused extra group
  __builtin_amdgcn_tensor_load_to_lds(g0, g1, g2, g3, gx, 0);
}
#endif

// ------------------------- small prologue kernels --------------------------

__global__ __launch_bounds__(256) void k_emb(const float* __restrict__ x,
                                             float* __restrict__ ws) {
  const int bc = blockIdx.x;           // b*64 + c
  const int t  = threadIdx.x;
  __shared__ float red[256];
  const float4* p4 = (const float4*)(x + (size_t)bc * HW_);
  float s = 0.f;
  for (int i = t; i < HW_ / 4; i += 256) {
    float4 v = p4[i];
    s += v.x + v.y + v.z + v.w;
  }
  red[t] = s;
  __syncthreads();
  for (int o = 128; o > 0; o >>= 1) {
    if (t < o) red[t] += red[t + o];
    __syncthreads();
  }
  if (t == 0) ws[WS_EMB + bc] = red[0] * (1.0f / (float)HW_);
}

__global__ __launch_bounds__(64) void k_pw(const float* __restrict__ w_lin,
                                           const float* __restrict__ b_lin,
                                           float* __restrict__ ws) {
  const int b = threadIdx.x;
  const float* eb = ws + WS_EMB + b * C_;
  float lg[P_];
#pragma unroll
  for (int p = 0; p < P_; ++p) {
    float s = b_lin[p];
    for (int c = 0; c < C_; ++c) s += eb[c] * w_lin[c * P_ + p];
    lg[p] = s;
  }
  float m = lg[0];
#pragma unroll
  for (int p = 1; p < P_; ++p) m = fmaxf(m, lg[p]);
  float se = 0.f;
#pragma unroll
  for (int p = 0; p < P_; ++p) { lg[p] = __expf(lg[p] - m); se += lg[p]; }
  float inv = 1.0f / se;
#pragma unroll
  for (int p = 0; p < P_; ++p) ws[WS_PW + b * P_ + p] = lg[p] * inv;
}

// ------------------------------ main kernel --------------------------------

__global__ __launch_bounds__(256) void k_main(
    const float* __restrict__ x, const float* __restrict__ sp,
    const float* __restrict__ wconv, const float* __restrict__ wgate,
    const float* __restrict__ w1, const float* __restrict__ b1,
    const float* __restrict__ w2, const float* __restrict__ b2,
    float* __restrict__ ws, float* __restrict__ out) {
  extern __shared__ float smem[];
  float* Xr  = smem + OFF_XR;
  float* Or  = smem + OFF_OR;
  float* WU  = smem + OFF_WU;      // W1 buffer
  float* WV  = smem + OFF_WV;      // WcT / W2 buffer
  float* H1  = smem + OFF_H1;
  float* Yl  = smem + OFF_Y;
  float* pr  = smem + OFF_PR;
  float* gts = smem + OFF_G;
  float* wg  = smem + OFF_WG;
  float* b1p = smem + OFF_B1;
  float* b2l = smem + OFF_B2;
  int*   cntI = (int*)(smem + OFF_CNT);
  int*   lstI = (int*)(smem + OFF_LST);

  const int t    = threadIdx.x;
  const int wv   = t >> 5;
  const int lane = t & 31;
  const int b    = blockIdx.x >> 5;
  const int s0   = (blockIdx.x & 31) * ST_;
  const int lm   = lane & 15;
  const int hih  = (lane >> 4);
  const int kofs = hih << 1;        // +0/+2 in K for A/B frags
  const int mofs = hih << 3;        // +0/+8 in M for C/D frags
  const int m0   = wv * 16;         // dense row tile for GEMM0

#if HAVE_TDM
  // kick off expert-0 W1 load immediately; it overlaps all the prologue work
  if (wv == 0)
    tdm_load_2d((unsigned)((uintptr_t)(void*)WU & 0xFFFFFFFFu), w1,
                HD_, C_, HDP_, C_, HD_);
#endif

  // ---- stage X tile (global [C][HW] -> LDS token-major [ST][C]) ----
  {
    const int c = t >> 2, q = t & 3;
    const float4* xrow = (const float4*)(x + ((size_t)(b * C_ + c)) * HW_ + s0);
#pragma unroll
    for (int i = 0; i < 8; ++i) {
      int j4 = q + 4 * i;
      float4 v = xrow[j4];
      int s = 4 * j4;
      Xr[(s + 0) * XS_ + c] = v.x;
      Xr[(s + 1) * XS_ + c] = v.y;
      Xr[(s + 2) * XS_ + c] = v.z;
      Xr[(s + 3) * XS_ + c] = v.w;
    }
  }
  // prompt values: pr[s] = sum_p pw[b,p]*sp[p, s0+s]
  if (t < ST_) {
    float acc = 0.f;
#pragma unroll
    for (int p = 0; p < P_; ++p)
      acc += ws[WS_PW + b * P_ + p] * sp[p * HW_ + s0 + t];
    pr[t] = acc;
  }
  // stage WcT into V: B[k=c][n=o] = wconv[o*64+c]
  for (int i = 0; i < 16; ++i) {
    int lin = t + i * 256;
    int o = lin >> 6, c = lin & 63;
    WV[c * C_ + o] = wconv[lin];
  }
  wg[t] = wgate[t];                 // 64x4 == 256
  // zero Y accumulator + routing lists
  for (int i = t; i < ST_ * XS_; i += 256) Yl[i] = 0.f;
  for (int i = t; i < E_ * ST_; i += 256) lstI[i] = 0;
  if (t < E_) cntI[t] = 0;
  if (t < 170) {                    // warm L2 ahead of the weight streams
    __builtin_prefetch(w1 + t * 64, 0, 1);
    __builtin_prefetch(w2 + t * 64, 0, 1);
  }
  __syncthreads();

  // ---- gating: top-2 of 4, softmax over the two, build compaction lists ----
  if (t < ST_) {
    const float* xrow = Xr + t * XS_;
    float l[E_] = {0.f, 0.f, 0.f, 0.f};
    for (int c = 0; c < C_; ++c) {
      float xv = xrow[c];
      const float* g4 = wg + c * E_;
#pragma unroll
      for (int e = 0; e < E_; ++e) l[e] += xv * g4[e];
    }
    int i1 = 0; float v1 = l[0];
#pragma unroll
    for (int e = 1; e < E_; ++e) if (l[e] > v1) { v1 = l[e]; i1 = e; }
    int i2 = -1; float v2 = -3.0e38f;
#pragma unroll
    for (int e = 0; e < E_; ++e)
      if (e != i1 && l[e] > v2) { v2 = l[e]; i2 = e; }
    float e2  = __expf(v2 - v1);
    float inv = 1.0f / (1.0f + e2);
#pragma unroll
    for (int e = 0; e < E_; ++e) gts[e * ST_ + t] = 0.f;
    gts[i1 * ST_ + t] = inv;
    gts[i2 * ST_ + t] = e2 * inv;
    int p1 = atomicAdd(&cntI[i1], 1);
    lstI[i1 * ST_ + p1] = t;
    int p2 = atomicAdd(&cntI[i2], 1);
    lstI[i2 * ST_ + p2] = t;
  }

  // ---- GEMM0 (dense): Out_r = (pr .* X_r)[128x64] x WcT[64x64] ----
  {
    const float prA = pr[m0 + lm];
    v8f acc[4];
#pragma unroll
    for (int nt = 0; nt < 4; ++nt) acc[nt] = (v8f)(0.0f);
    for (int k = 0; k < C_; k += 4) {
      const float* ap = Xr + (m0 + lm) * XS_ + k + kofs;
      v2f a; a[0] = prA * ap[0]; a[1] = prA * ap[1];
#pragma unroll
      for (int nt = 0; nt < 4; ++nt) {
        const float* bp = WV + (k + kofs) * C_ + nt * 16 + lm;
        v2f bb; bb[0] = bp[0]; bb[1] = bp[C_];
        acc[nt] = wmma4(a, bb, acc[nt]);
      }
    }
#pragma unroll
    for (int nt = 0; nt < 4; ++nt)
#pragma unroll
      for (int v = 0; v < 8; ++v)
        Or[(m0 + v + mofs) * XS_ + nt * 16 + lm] = acc[nt][v];
  }
  __syncthreads();

  // deterministic per-block gating stats (fixed summation order)
  if (t < 4) {
    float imp = 0.f; float ld = 0.f;
    for (int s = 0; s < ST_; ++s) {
      float g = gts[t * ST_ + s];
      imp += g;
      ld  += (g > 0.f) ? 1.0f : 0.0f;
    }
    ws[WS_PART + (size_t)blockIdx.x * 8 + t]     = imp;
    ws[WS_PART + (size_t)blockIdx.x * 8 + 4 + t] = ld;
  }

  // ---- expert loop over compacted token lists ----
  for (int e = 0; e < E_; ++e) {
#if HAVE_TDM
    // V is free (consumed last iteration / by GEMM0): stream W2(e) into it
    if (wv == 0)
      tdm_load_2d((unsigned)((uintptr_t)(void*)WV & 0xFFFFFFFFu),
                  w2 + (size_t)e * HD_ * C_, C_, HD_, C_, HDP_, C_);
#endif
    // biases (tiny, all threads)
    if (t < HDP_) b1p[t] = (t < HD_) ? b1[e * HD_ + t] : 0.0f;
    if (t < C_)   b2l[t] = b2[e * C_ + t];
    if (e + 1 < E_ && t < 170)
      __builtin_prefetch(w1 + (size_t)(e + 1) * C_ * HD_ + t * 64, 0, 1);
#if HAVE_TDM
    if (wv == 0) __builtin_amdgcn_s_wait_tensorcnt(1);   // W1(e) has landed
#else
    {   // manual stage W1(e) -> WU[c][176], zero-pad hd 170..175
      const int c = t >> 2, part = t & 3;
      const float* src = w1 + ((size_t)(e * C_ + c)) * HD_;
      float* dst = WU + c * HDP_;
#pragma unroll
      for (int j = 0; j < 44; ++j) {
        int hd = part * 44 + j;
        dst[hd] = (hd < HD_) ? src[hd] : 0.0f;
      }
    }
#endif
    __syncthreads();

    const int  cnt   = __builtin_amdgcn_readfirstlane(cntI[e]);
    const int  ntile = (cnt + 15) >> 4;            // compacted 16-row tiles
    const int* lst   = lstI + e * ST_;

    // GEMM1 (compacted): H1[r] = gate(tok) * gelu(Or[tok] x W1 + b1)
    // jobs = ntile x 11 over 8 waves; padded rows route to token 0 (ignored).
    for (int j = wv; j < ntile * 11; j += 8) {
      int mt = j / 11, nt = j - mt * 11;
      int ra = lst[mt * 16 + lm];                  // A-row gather index
      v8f acc = (v8f)(0.0f);
      for (int k = 0; k < C_; k += 4) {
        const float* ap = Or + ra * XS_ + k + kofs;
        v2f a; a[0] = ap[0]; a[1] = ap[1];
        const float* bp = WU + (k + kofs) * HDP_ + nt * 16 + lm;
        v2f bb; bb[0] = bp[0]; bb[1] = bp[HDP_];
        acc = wmma4(a, bb, acc);
      }
      const int n = nt * 16 + lm;
      const float b1v = b1p[n];
#pragma unroll
      for (int v = 0; v < 8; ++v) {
        int rr  = mt * 16 + v + mofs;
        int tok = lst[rr];
        float g = gts[e * ST_ + tok];
        H1[rr * HS_ + n] = g * gelu_tanh(acc[v] + b1v);
      }
    }
    __syncthreads();     // H1 complete; all waves done reading WU

#if HAVE_TDM
    if (wv == 0) {
      if (e + 1 < E_) {   // stream next expert's W1 while GEMM2 runs
        tdm_load_2d((unsigned)((uintptr_t)(void*)WU & 0xFFFFFFFFu),
                    w1 + (size_t)(e + 1) * C_ * HD_, HD_, C_, HDP_, C_, HD_);
        __builtin_amdgcn_s_wait_tensorcnt(1);   // W2(e) has landed
      } else {
        __builtin_amdgcn_s_wait_tensorcnt(0);
      }
    }
#else
    for (int i = 0; i < 44; ++i) {   // manual stage W2(e) -> WV[hd][64]
      int lin = t + i * 256;
      int hd = lin >> 6;
      WV[lin] = (hd < HD_) ? w2[(size_t)e * HD_ * C_ + lin] : 0.0f;
    }
    if (e + 1 < E_ && t < 170)
      __builtin_prefetch(w2 + (size_t)(e + 1) * HD_ * C_ + t * 64, 0, 1);
#endif
    __syncthreads();

    // GEMM2 (compacted): Y[tok] += H1c x W2 + gate*b2; scatter through list.
    // Each token appears at most once per expert -> race-free LDS adds.
    for (int j = wv; j < ntile * 4; j += 8) {
      int mt = j >> 2, nt = j & 3;
      v8f acc = (v8f)(0.0f);
      for (int k = 0; k < HDP_; k += 4) {
        const float* ap = H1 + (mt * 16 + lm) * HS_ + k + kofs;
        v2f a; a[0] = ap[0]; a[1] = ap[1];
        const float* bp = WV + (k + kofs) * C_ + nt * 16 + lm;
        v2f bb; bb[0] = bp[0]; bb[1] = bp[C_];
        acc = wmma4(a, bb, acc);
      }
      const int c = nt * 16 + lm;
      const float b2v = b2l[c];
#pragma unroll
      for (int v = 0; v < 8; ++v) {
        int rr = mt * 16 + v + mofs;
        if (rr < cnt) {
          int tok = lst[rr];
          float g = gts[e * ST_ + tok];
          Yl[tok * XS_ + c] += acc[v] + g * b2v;
        }
      }
    }
    __syncthreads();   // done reading H1/WV/Y before next expert overwrites
  }

  // ---- epilogue: out = Y + x, back to [C][HW] layout ----
  {
    const int c0 = t >> 5, j = t & 31;
#pragma unroll
    for (int r = 0; r < 8; ++r) {
      int cc = c0 + r * 8;
      float4 o;
      int s = 4 * j;
      o.x = Yl[(s + 0) * XS_ + cc] + Xr[(s + 0) * XS_ + cc];
      o.y = Yl[(s + 1) * XS_ + cc] + Xr[(s + 1) * XS_ + cc];
      o.z = Yl[(s + 2) * XS_ + cc] + Xr[(s + 2) * XS_ + cc];
      o.w = Yl[(s + 3) * XS_ + cc] + Xr[(s + 3) * XS_ + cc];
      ((float4*)(out + ((size_t)(b * C_ + cc)) * HW_ + s0))[j] = o;
    }
  }
}

// ------------------------------ loss kernel --------------------------------

__global__ __launch_bounds__(32) void k_loss(const float* __restrict__ ws,
                                             float* __restrict__ out) {
  __shared__ float tot[8];
  const int t = threadIdx.x;
  if (t < 8) {
    float s = 0.f;
    for (int blk = 0; blk < NBLK; ++blk)          // fixed order: deterministic
      s += ws[WS_PART + (size_t)blk * 8 + t];
    tot[t] = s;
  }
  __syncthreads();
  if (t == 0) {
    float mi = 0.f, qi = 0.f, ml = 0.f, ql = 0.f;
#pragma unroll
    for (int e = 0; e < E_; ++e) {
      float v = tot[e];     mi += v; qi += v * v;
      float u = tot[4 + e]; ml += u; ql += u * u;
    }
    mi *= 0.25f; qi *= 0.25f; ml *= 0.25f; ql *= 0.25f;
    float vari = qi - mi * mi, varl = ql - ml * ml;
    out[(size_t)B_ * C_ * HW_] =
        vari / (mi * mi + 1e-10f) + varl / (ml * ml + 1e-10f);
  }
}

// ------------------------------- launcher ----------------------------------

extern "C" void kernel_launch(void* const* d_in, const int* in_sizes, int n_in,
                              void* d_out, int out_size, void* d_ws,
                              size_t ws_size, hipStream_t stream) {
  const float* x     = (const float*)d_in[0];
  // d_in[1] = text (unused by the reference)
  const float* sp    = (const float*)d_in[2];
  const float* w_lin = (const float*)d_in[3];
  const float* b_lin = (const float*)d_in[4];
  const float* wconv = (const float*)d_in[5];
  const float* wgate = (const float*)d_in[6];
  const float* w1    = (const float*)d_in[7];
  const float* b1    = (const float*)d_in[8];
  const float* w2    = (const float*)d_in[9];
  const float* b2    = (const float*)d_in[10];
  float* out = (float*)d_out;
  float* ws  = (float*)d_ws;

  k_emb<<<B_ * C_, 256, 0, stream>>>(x, ws);
  k_pw<<<1, 64, 0, stream>>>(w_lin, b_lin, ws);
  k_main<<<NBLK, 256, LDS_BYTES, stream>>>(x, sp, wconv, wgate, w1, b1, w2, b2,
                                           ws, out);
  k_loss<<<1, 32, 0, stream>>>(ws, out);
}